// RBF_23888608100601
// MI455X (gfx1250) — compile-verified
//
#include <hip/hip_runtime.h>
#include <math.h>

#define F 256
#define EPSF 2.220446049250313e-16f

typedef float v2f __attribute__((ext_vector_type(2)));
typedef float v8f __attribute__((ext_vector_type(8)));

// ---------------------------------------------------------------------------
// Stage 1: psi row sums.  psi[j][k] = (A[k][j]+eps)*(A[j][k]+eps)  (symmetric)
// g[j] = rowsum_j + colsum_j = 2*rowsum_j ;  S0 = sum_jk psi
// ---------------------------------------------------------------------------
__global__ void __launch_bounds__(256)
rbf_psi_sums(const float* __restrict__ A, float* __restrict__ g,
             float* __restrict__ S0) {
    __shared__ float red[256];
    const int j = threadIdx.x;
    const float* Arow = A + (size_t)j * F;   // A[j][k]
    const float* Acol = A + j;               // A[k][j]
    float rs = 0.f;
    for (int k = 0; k < F; ++k) {
        rs += (Acol[(size_t)k * F] + EPSF) * (Arow[k] + EPSF);
    }
    g[j]   = 2.0f * rs;
    red[j] = rs;
    __syncthreads();
    for (int s = 128; s > 0; s >>= 1) {
        if (j < s) red[j] += red[j + s];
        __syncthreads();
    }
    if (j == 0) S0[0] = red[0];
}

// ---------------------------------------------------------------------------
// Stage 2: per-feature constants for center row i:
//   S1[i] = sum_j center[i][j] * g[j]
//   S2[i] = sum_jk psi[j][k] * center[i][j] * center[i][k]
// One block per i, thread j owns row j of psi (recomputed on the fly).
// ---------------------------------------------------------------------------
__global__ void __launch_bounds__(256)
rbf_center_consts(const float* __restrict__ A, const float* __restrict__ center,
                  const float* __restrict__ g, float* __restrict__ S1,
                  float* __restrict__ S2) {
    __shared__ float cs[F];
    __shared__ float red1[256];
    __shared__ float red2[256];
    const int i = blockIdx.x;
    const int j = threadIdx.x;
    const float cij = center[(size_t)i * F + j];
    cs[j] = cij;
    __syncthreads();
    const float* Arow = A + (size_t)j * F;
    const float* Acol = A + j;
    float inner = 0.f;
    for (int k = 0; k < F; ++k) {
        float psi = (Acol[(size_t)k * F] + EPSF) * (Arow[k] + EPSF);
        inner += psi * cs[k];
    }
    red1[j] = cij * g[j];
    red2[j] = cij * inner;
    __syncthreads();
    for (int s = 128; s > 0; s >>= 1) {
        if (j < s) { red1[j] += red1[j + s]; red2[j] += red2[j + s]; }
        __syncthreads();
    }
    if (j == 0) { S1[i] = red1[0]; S2[i] = red2[0]; }
}

// ---------------------------------------------------------------------------
// Stage 3: fused activation + GEMM.
//   M[r][k] = exp(-beta_k * sqrt(S0*x^2 - S1_k*x + S2_k)),  x = X[r][k]
//   out     = M @ W + bias   via V_WMMA_F32_16X16X4_F32
// Block: 256 threads (8 waves), 32-row strip.  Wave w: rows (w&1)*16..+15,
// cols (w>>1)*64..+63 (4 accumulator tiles of 16x16).
// ---------------------------------------------------------------------------
__global__ void __launch_bounds__(256)
rbf_wmma_gemm(const float* __restrict__ X, const float* __restrict__ betas,
              const float* __restrict__ W, const float* __restrict__ bias,
              const float* __restrict__ S1, const float* __restrict__ S2,
              const float* __restrict__ S0p, float* __restrict__ out) {
    constexpr int ROWS = 32;
    constexpr int LDA  = F + 4;                 // 260 floats: conflict-free, 8B-aligned rows
    __shared__ float As[ROWS * LDA];

    const int tid = threadIdx.x;
    const int r0  = blockIdx.x * ROWS;
    const float S0 = S0p[0];

    // --- fused mahalanobis tile into LDS (coalesced on k) ---
    for (int e = tid; e < ROWS * F; e += 256) {
        const int row = e >> 8;                 // F == 256
        const int k   = e & (F - 1);
        const float x = X[(size_t)(r0 + row) * F + k];
        const float quad = S0 * x * x - S1[k] * x + S2[k];
        const float m = expf(-betas[k] * sqrtf(quad));
        As[row * LDA + k] = m;
    }
    __syncthreads();

    const int lane     = tid & 31;
    const int wid      = tid >> 5;
    const int row_base = (wid & 1) * 16;
    const int n_base   = (wid >> 1) * 64;
    const int lhalf    = lane >> 4;             // 0: K=0,1   1: K=2,3
    const int lmod     = lane & 15;

    v8f acc[4];
#pragma unroll
    for (int t = 0; t < 4; ++t) acc[t] = (v8f){0.f,0.f,0.f,0.f,0.f,0.f,0.f,0.f};

    const float* As_lane = &As[(row_base + lmod) * LDA];

    for (int k = 0; k < F; k += 4) {
        const int kk = k + 2 * lhalf;
        // A fragment: 16x4 f32 — lane holds (M=lmod, K=kk) and (M=lmod, K=kk+1)
        const v2f a = *(const v2f*)(As_lane + kk);
#pragma unroll
        for (int t = 0; t < 4; ++t) {
            const int n = n_base + 16 * t + lmod;
            // B fragment: 4x16 f32 — lane holds (K=kk, N=n) and (K=kk+1, N=n)
            v2f b;
            b.x = W[(size_t)kk * F + n];
            b.y = W[(size_t)(kk + 1) * F + n];
            acc[t] = __builtin_amdgcn_wmma_f32_16x16x4_f32(
                false, a, false, b, (short)0, acc[t], false, false);
        }
    }

    // --- epilogue: D layout — VGPR v: lanes 0-15 -> M=v, lanes 16-31 -> M=v+8 ---
#pragma unroll
    for (int t = 0; t < 4; ++t) {
        const int col = n_base + 16 * t + lmod;
        const float bv = bias[col];
#pragma unroll
        for (int v = 0; v < 8; ++v) {
            const int row = r0 + row_base + v + 8 * lhalf;
            out[(size_t)row * F + col] = acc[t][v] + bv;
        }
    }
}

// ---------------------------------------------------------------------------
extern "C" void kernel_launch(void* const* d_in, const int* in_sizes, int n_in,
                              void* d_out, int out_size, void* d_ws, size_t ws_size,
                              hipStream_t stream) {
    const float* x      = (const float*)d_in[0];
    const float* betas  = (const float*)d_in[1];
    const float* center = (const float*)d_in[2];
    const float* A      = (const float*)d_in[3];
    const float* weight = (const float*)d_in[4];
    const float* bias   = (const float*)d_in[5];
    float* out = (float*)d_out;

    // ws layout: g[F] | S1[F] | S2[F] | S0[1]
    float* g  = (float*)d_ws;
    float* S1 = g + F;
    float* S2 = S1 + F;
    float* S0 = S2 + F;

    const int R = in_sizes[0] / F;   // B*T = 4096

    rbf_psi_sums<<<1, 256, 0, stream>>>(A, g, S0);
    rbf_center_consts<<<F, 256, 0, stream>>>(A, center, g, S1, S2);
    rbf_wmma_gemm<<<R / 32, 256, 0, stream>>>(x, betas, weight, bias, S1, S2, S0, out);
}